// MetaLearner_25056839205556
// MI455X (gfx1250) — compile-verified
//
#include <hip/hip_runtime.h>

// Meta-learner fused update:  out = p + (w_g*g + w_p*p)  over 6 fp32 tensors.
// Pure streaming / HBM-bound (AI = 0.25 FLOP/byte, ~541 MB, floor ~23 us at
// 23.3 TB/s). Optimized for MI455X with B128 vmem, NT temporal hints, and
// gfx1250 global_prefetch. WMMA is intentionally NOT used: matrix ops cannot
// reduce byte traffic for an elementwise op and would only add layout cost.

typedef float v4f __attribute__((ext_vector_type(4)));

#define NTEN  6                 // number of parameter tensors
#define TPB   256               // 8 wave32 per block
#define ITEMS 4                 // float4s per thread
#define TILE  (TPB * ITEMS)     // float4s per block (1024 -> 16 KB out/block)
#define PFD   (8 * TILE)        // prefetch distance in float4s (~128 KB ahead)

struct Desc {
  const v4f*  p[NTEN];
  const v4f*  g[NTEN];
  v4f*        o[NTEN];          // d_out + cumulative offset per tensor
  const float* w;               // [w_g, w_p]
  int blk_start[NTEN + 1];      // cumulative block starts per tensor
  int n4[NTEN];                 // float4 count per tensor
};

__global__ __launch_bounds__(TPB) void meta_update_kernel(Desc d) {
  // ---- Uniform per-block tensor lookup: blockIdx-derived -> pure SALU ----
  const int b = (int)blockIdx.x;
  int t = 0;
#pragma unroll
  for (int i = 1; i < NTEN; ++i) t += (b >= d.blk_start[i]) ? 1 : 0;

  const int lb   = b - d.blk_start[t];
  const int n4   = d.n4[t];
  const int base = lb * TILE + (int)threadIdx.x;

  // Uniform address -> scalar loads (s_load), broadcast to all lanes.
  const float wg = d.w[0];
  const float wp = d.w[1];

  const v4f* __restrict__ P = d.p[t];
  const v4f* __restrict__ G = d.g[t];
  v4f*       __restrict__ O = d.o[t];

#pragma unroll
  for (int j = 0; j < ITEMS; ++j) {
    const int idx = base + j * TPB;
    if (idx < n4) {
      // gfx1250 streaming prefetch of the stream ahead (global_prefetch_b8).
      const int pf = idx + PFD;
      if (pf < n4) {
        __builtin_prefetch((const void*)(P + pf), 0, 0);
        __builtin_prefetch((const void*)(G + pf), 0, 0);
      }
      // Stream-once data: non-temporal B128 loads/stores (TH=NT) so the
      // 541 MB stream doesn't thrash the 192 MB L2.
      v4f pv = __builtin_nontemporal_load(P + idx);
      v4f gv = __builtin_nontemporal_load(G + idx);
      v4f r  = pv + (wg * gv + wp * pv);   // p + (w_g*g + w_p*p)
      __builtin_nontemporal_store(r, O + idx);
    }
  }
}

extern "C" void kernel_launch(void* const* d_in, const int* in_sizes, int n_in,
                              void* d_out, int out_size, void* d_ws, size_t ws_size,
                              hipStream_t stream) {
  (void)n_in; (void)out_size; (void)d_ws; (void)ws_size;

  // Input order (setup_inputs dict): weights, p0,g0, p1,g1, ..., p5,g5.
  Desc d;
  d.w = (const float*)d_in[0];

  float* out = (float*)d_out;
  long  off = 0;   // running element offset into concatenated output
  int   blk = 0;   // running block offset

  for (int t = 0; t < NTEN; ++t) {
    const int n = in_sizes[1 + 2 * t];          // element count of p_t (== g_t)
    d.p[t] = (const v4f*)d_in[1 + 2 * t];
    d.g[t] = (const v4f*)d_in[2 + 2 * t];
    d.o[t] = (v4f*)(out + off);                 // off is a multiple of 4 -> 16B aligned
    const int n4 = n / 4;                       // all SHAPES are multiples of 4 floats
    d.n4[t]        = n4;
    d.blk_start[t] = blk;
    blk += (n4 + TILE - 1) / TILE;
    off += n;
  }
  d.blk_start[NTEN] = blk;

  meta_update_kernel<<<dim3((unsigned)blk), dim3(TPB), 0, stream>>>(d);
}